// Convolution2D_1812476199028
// MI455X (gfx1250) — compile-verified
//
#include <hip/hip_runtime.h>

typedef __attribute__((ext_vector_type(2))) float v2f;
typedef __attribute__((ext_vector_type(8))) float v8f;

#if defined(__has_builtin)
#  if __has_builtin(__builtin_amdgcn_global_load_async_to_lds_b32)
#    define USE_ASYNC_LDS 1
#  endif
#endif

namespace {
constexpr int HH   = 255;
constexpr int WW   = 255;
constexpr int NPIX = HH * WW;   // 65025 (odd -> exact median is order stat 32512, 0-indexed)
constexpr int CH   = 32;
constexpr int NB   = 8;
constexpr int QQ   = 4;
}

// Monotone bijection float -> uint so radix select gives the float order statistic.
__device__ __forceinline__ unsigned f2key(float f) {
  unsigned u = __float_as_uint(f);
  return (u & 0x80000000u) ? ~u : (u | 0x80000000u);
}
__device__ __forceinline__ float key2f(unsigned k) {
  unsigned u = (k & 0x80000000u) ? (k & 0x7fffffffu) : ~k;
  return __uint_as_float(u);
}

// One workgroup per (b,c) plane. 4-pass 8-bit radix select, 256-bin LDS histogram.
// Plane is 255KB -> L2 resident after first pass; deterministic (atomics are counters only).
__global__ __launch_bounds__(256) void median_radix_kernel(const float* __restrict__ x,
                                                           float* __restrict__ med) {
  __shared__ unsigned hist[256];
  __shared__ unsigned s_rank, s_prefix;
  const int tid = threadIdx.x;
  const int bc  = blockIdx.x;
  const float* plane = x + (size_t)bc * NPIX;

  if (tid == 0) { s_rank = (NPIX - 1) / 2; s_prefix = 0u; }
  unsigned prefix = 0u;
  for (int shift = 24; shift >= 0; shift -= 8) {
    hist[tid] = 0u;
    __syncthreads();
    const unsigned mask = (shift == 24) ? 0u : (0xFFFFFFFFu << (shift + 8));
    for (int i = tid; i < NPIX; i += 256) {
      unsigned k = f2key(plane[i]);
      if ((k & mask) == prefix) atomicAdd(&hist[(k >> shift) & 0xFFu], 1u);
    }
    __syncthreads();
    if (tid == 0) {
      unsigned r = s_rank;
      unsigned b = 0;
      for (b = 0; b < 256u; ++b) {
        unsigned h = hist[b];
        if (r < h) break;
        r -= h;
      }
      s_rank   = r;
      s_prefix = prefix | (b << shift);
    }
    __syncthreads();
    prefix = s_prefix;
  }
  if (tid == 0) med[bc] = key2f(prefix);
}

// One workgroup (8 waves) per (b, c, output row). Stage 3 median-padded input rows in
// LDS (async-to-LDS on CDNA5), then each wave computes a 32-pixel strip as two 16-pixel
// WMMA tiles:
//   D[16x16] = A(16x4: rows 0..3 = W chunk) x B(4x16: patch values per pixel) + C(bias)
// p in [0,12) split into 3 K=4 chunks (p=9..11 zero-padded).
__global__ __launch_bounds__(256) void conv_wmma_kernel(const float* __restrict__ x,
                                                        const float* __restrict__ Wm,
                                                        const float* __restrict__ bias,
                                                        const float* __restrict__ med,
                                                        float* __restrict__ out) {
  __shared__ float srow[3][264];          // srow[r][i] = padded(y-1+r, i-1); i in [0,256]

  const int y    = blockIdx.x;            // 0..254
  const int cc   = blockIdx.y;            // 0..31
  const int b    = blockIdx.z;            // 0..7
  const int tid  = threadIdx.x;
  const int lane = tid & 31;
  const int wave = tid >> 5;              // 0..7

  const float m = med[b * CH + cc];
  const float* plane = x + (size_t)(b * CH + cc) * NPIX;

  // Cooperative halo stage: 3 padded rows x 257 cols; border == median.
  for (int idx = tid; idx < 3 * 257; idx += 256) {
    const int r  = idx / 257;
    const int i  = idx - r * 257;         // padded col; real xcoord = i-1
    const int yy = y - 1 + r;
    const int xx = i - 1;
    const bool inb = (yy >= 0) && (yy < HH) && (xx >= 0) && (xx < WW);
#if defined(USE_ASYNC_LDS)
    if (inb) {
      // CDNA5 async copy: HBM/L2 -> LDS directly, tracked by ASYNCcnt (no VGPR round trip).
      __builtin_amdgcn_global_load_async_to_lds_b32(
          (__attribute__((address_space(1))) int*)(plane + yy * WW + xx),
          (__attribute__((address_space(3))) int*)&srow[r][i],
          /*imm offset*/ 0, /*cpol*/ 0);
    } else {
      srow[r][i] = m;                     // disjoint addresses: safe alongside async writes
    }
#else
    srow[r][i] = inb ? plane[yy * WW + xx] : m;
#endif
  }

  // --- A fragments (ISA 16x4 f32 layout: lanes 0-15 hold K=0,1; lanes 16-31 hold K=2,3)
  const int mrow  = lane & 15;            // M index (only 0..3 used)
  const int kbase = (lane >> 4) * 2;      // K pair base for this half-wave
  v2f a[3];
#pragma unroll
  for (int kc = 0; kc < 3; ++kc) {
    const int p0 = kc * 4 + kbase;
    const int p1 = p0 + 1;
    const bool v0 = (mrow < QQ) && (p0 < 9);
    const bool v1 = (mrow < QQ) && (p1 < 9);
    float a0 = Wm[v0 ? (mrow * 9 + p0) : 0];
    float a1 = Wm[v1 ? (mrow * 9 + p1) : 0];
    a[kc].x = v0 ? a0 : 0.0f;
    a[kc].y = v1 ? a1 : 0.0f;
  }

  // --- C seeded with bias: D/C layout -> VGPR r, lanes 0-15 = row M=r (rows 4..15 unused)
  v8f acc0 = {};
  v8f acc1 = {};
#pragma unroll
  for (int r = 0; r < 8; ++r) {
    const float bb = bias[r & 3];
    const float ci = (r < QQ && lane < 16) ? bb : 0.0f;
    acc0[r] = ci;
    acc1[r] = ci;
  }

#if defined(USE_ASYNC_LDS)
  asm volatile("s_wait_asynccnt 0x0" ::: "memory");  // this wave's LDS writes landed
#endif
  __syncthreads();

  // --- B fragments from LDS + 3 chained WMMAs per 16-pixel strip
  const int xbase = wave * 32;
  const int n = lane & 15;                // N (pixel) index within tile
#pragma unroll
  for (int kc = 0; kc < 3; ++kc) {
    const int p0 = kc * 4 + kbase;
    const int p1 = p0 + 1;
    const int x0 = xbase + n;             // strip 0 pixel
    const int x1 = xbase + 16 + n;        // strip 1 pixel
    v2f b0, b1;
    b0.x = (p0 < 9) ? srow[p0 / 3][x0 + p0 % 3] : 0.0f;
    b0.y = (p1 < 9) ? srow[p1 / 3][x0 + p1 % 3] : 0.0f;
    b1.x = (p0 < 9) ? srow[p0 / 3][x1 + p0 % 3] : 0.0f;
    b1.y = (p1 < 9) ? srow[p1 / 3][x1 + p1 % 3] : 0.0f;
    acc0 = __builtin_amdgcn_wmma_f32_16x16x4_f32(false, a[kc], false, b0,
                                                 (short)0, acc0, false, false);
    acc1 = __builtin_amdgcn_wmma_f32_16x16x4_f32(false, a[kc], false, b1,
                                                 (short)0, acc1, false, false);
  }

  // --- Store: merge strip1 into lanes 16-31 via ds_swizzle SWAPX16, then non-temporal
  // 128B-contiguous stores (266MB write stream must not evict x from L2).
  const int xcol = xbase + lane;
#pragma unroll
  for (int q = 0; q < QQ; ++q) {
    const float sw = __int_as_float(
        __builtin_amdgcn_ds_swizzle(__float_as_int(acc1[q]), 0x401f));  // SWAPX16
    const float v = (lane < 16) ? acc0[q] : sw;
    if (xcol < WW) {
      __builtin_nontemporal_store(
          v, &out[(((size_t)b * (QQ * CH) + (size_t)q * CH + cc) * HH + y) * WW + xcol]);
    }
  }
}

extern "C" void kernel_launch(void* const* d_in, const int* in_sizes, int n_in,
                              void* d_out, int out_size, void* d_ws, size_t ws_size,
                              hipStream_t stream) {
  (void)in_sizes; (void)n_in; (void)out_size; (void)ws_size;
  const float* x  = (const float*)d_in[0];   // (8, 32, 255, 255) fp32
  const float* Wm = (const float*)d_in[1];   // (4, 9) fp32
  const float* bb = (const float*)d_in[2];   // (4,) fp32
  float* out = (float*)d_out;                // (8, 128, 255, 255) fp32
  float* med = (float*)d_ws;                 // 256 per-(b,c) medians

  median_radix_kernel<<<NB * CH, 256, 0, stream>>>(x, med);
  conv_wmma_kernel<<<dim3(HH, CH, NB), 256, 0, stream>>>(x, Wm, bb, med, out);
}